// FCOSPostProcessor_33956011442610
// MI455X (gfx1250) — compile-verified
//
#include <hip/hip_runtime.h>
#include <hip/hip_bf16.h>
#include <cstdint>

// ---------------------------------------------------------------------------
// FCOS post-processor for MI455X (gfx1250).
// Bandwidth-bound: 67MB of cls logits dominate. Data path: TDM async loads to
// LDS (tensor_load_to_lds + s_wait_tensorcnt), double-buffered, wave32 blocks.
// ---------------------------------------------------------------------------

typedef unsigned int v4u __attribute__((ext_vector_type(4)));
typedef int          v8i __attribute__((ext_vector_type(8)));
typedef int          v4i __attribute__((ext_vector_type(4)));

#define BATCH   16
#define NC      80
#define N_L0    10000
#define N_L1    2500
#define N_L2    625
#define KTOP    1000
#define CAP     8192
#define TOTC    3000          // 3 levels * KTOP
#define NPAD2   4096
#define POSTN   100
#define BINS    4096
#define STAGE   4096          // floats per TDM stage (16 KB)
#define CHUNK   (STAGE * 8)   // floats per block
#define IMGMAX  799.0f
#define LOGIT_THRESH (-2.9444389791664403f)   // logit(0.05)
#define PRE_THRESH   0.05f
#define NMS_IOU      0.6f

static constexpr int CH0   = (NC * N_L0 + CHUNK - 1) / CHUNK;  // 25
static constexpr int CH1   = (NC * N_L1 + CHUNK - 1) / CHUNK;  // 7
static constexpr int CH2   = (NC * N_L2 + CHUNK - 1) / CHUNK;  // 2
static constexpr int CHTOT = CH0 + CH1 + CH2;                  // 34

// workspace layout in u32 units
static constexpr size_t WS_HIST = 0;
static constexpr size_t WS_CUT  = WS_HIST + (size_t)BATCH * 3 * BINS;
static constexpr size_t WS_CNT  = WS_CUT + BATCH * 3;
static constexpr size_t WS_CAND = WS_CNT + BATCH * 3;
static constexpr size_t WS_SLAB = WS_CAND + (size_t)BATCH * 3 * CAP * 2;
static constexpr int    WS_ZERO_WORDS = (int)WS_CAND;

#if defined(__HIP_DEVICE_COMPILE__) && __has_builtin(__builtin_amdgcn_tensor_load_to_lds)
#define HAVE_TDM 1
#else
#define HAVE_TDM 0
#endif

// Stage `n` f32 elements from global `g` into LDS double-buffer slot `bufIdx`.
// TDM path: 1-D D# descriptor, data_size=4B, tile_dim0=n, tile_dim1=1.
// lds_addr assumes the single static __shared__ block starts at LDS offset 0.
__device__ __forceinline__ void stage_fill(unsigned* lds_u32, int bufIdx,
                                           const float* g, int n, int lane) {
#if HAVE_TDM
  (void)lds_u32; (void)lane;
  unsigned long long ga = (unsigned long long)(uintptr_t)g;
  unsigned ne = (unsigned)n;
  v4u g0;
  g0.x = 1u;                                              // count=1, user D#
  g0.y = (unsigned)(bufIdx * STAGE * 4);                  // lds_addr [63:32]
  g0.z = (unsigned)(ga & 0xFFFFFFFFull);                  // global_addr[31:0]
  g0.w = (unsigned)((ga >> 32) & 0x01FFFFFFull) | (2u << 30); // ga[56:32], type=2
  v8i g1 = {
    (int)(2u << 16),                                      // data_size = 4 bytes
    (int)((ne & 0xFFFFu) << 16),                          // tensor_dim0[15:0]
    (int)(((ne >> 16) & 0xFFFFu) | (1u << 16)),           // tensor_dim0[31:16], tensor_dim1=1
    (int)((ne & 0xFFFFu) << 16),                          // tile_dim0 = ne
    1,                                                    // tile_dim1 = 1
    (int)ne,                                              // tensor_dim0_stride low32
    0, 0
  };
  v4i z4 = {0, 0, 0, 0};
  v8i z8 = {0, 0, 0, 0, 0, 0, 0, 0};
  // clang-23 / therock-10.0 6-arg form: (g0, g1, g2, g3, g4, cpol)
  __builtin_amdgcn_tensor_load_to_lds(g0, g1, z4, z4, z8, 0);
#else
  float* dst = (float*)&lds_u32[(size_t)bufIdx * STAGE];
  for (int i = lane; i < n; i += 32) dst[i] = g[i];
#endif
}

__device__ __forceinline__ float sigmoidf_fast(float x) {
  return 1.0f / (1.0f + __expf(-x));
}

// -------------------------- kernel 0: zero scratch --------------------------
__global__ void zero_ws_kernel(unsigned* ws, int nwords) {
  for (int i = blockIdx.x * blockDim.x + threadIdx.x; i < nwords;
       i += gridDim.x * blockDim.x)
    ws[i] = 0u;
}

// ------------------- kernels 1 & 3: TDM-staged logit scan -------------------
// MODE 0: build per-(image,level) 4096-bin histogram of score float bits.
// MODE 1: compact candidates with bin >= cutoff into cand[] (atomic counter).
template <int MODE>
__global__ __launch_bounds__(32) void scan_kernel(
    const float* __restrict__ cls0, const float* __restrict__ cls1,
    const float* __restrict__ cls2, const float* __restrict__ cn0,
    const float* __restrict__ cn1, const float* __restrict__ cn2,
    unsigned* __restrict__ hist, const unsigned* __restrict__ cut,
    unsigned* __restrict__ cnt, unsigned* __restrict__ cand) {
  __shared__ unsigned lds[STAGE * 2 + BINS];  // 2 stage buffers + LDS histogram
  const int lane = threadIdx.x;

  int bid = blockIdx.x;
  int b = bid / CHTOT;
  int r = bid % CHTOT;
  int level, chunk;
  if (r < CH0)            { level = 0; chunk = r; }
  else if (r < CH0 + CH1) { level = 1; chunk = r - CH0; }
  else                    { level = 2; chunk = r - CH0 - CH1; }

  const float* clsP = (level == 0) ? cls0 : (level == 1) ? cls1 : cls2;
  const float* cnP  = (level == 0) ? cn0  : (level == 1) ? cn1  : cn2;
  const int N       = (level == 0) ? N_L0 : (level == 1) ? N_L1 : N_L2;
  const int total   = NC * N;
  const int chunkBase = chunk * CHUNK;
  const int nElems = min(CHUNK, total - chunkBase);
  if (nElems <= 0) return;
  const float* src = clsP + (size_t)b * total + chunkBase;

  if (MODE == 0)
    for (int i = lane; i < BINS; i += 32) lds[STAGE * 2 + i] = 0u;

  unsigned cutBin = 0;
  if (MODE == 1) cutBin = cut[b * 3 + level];

  const int nStages = (nElems + STAGE - 1) / STAGE;
  stage_fill(lds, 0, src, min(STAGE, nElems), lane);          // prime stage 0
  for (int s = 0; s < nStages; ++s) {
    const int sBase = s * STAGE;
    const int sCount = min(STAGE, nElems - sBase);
    if (s + 1 < nStages) {
      const int nBase = sBase + STAGE;
      asm volatile("s_wait_dscnt 0" ::: "memory");  // buffer-reuse guard
      stage_fill(lds, (s + 1) & 1, src + nBase, min(STAGE, nElems - nBase), lane);
      __builtin_amdgcn_s_wait_tensorcnt(1);         // stage s landed (in-order)
    } else {
      __builtin_amdgcn_s_wait_tensorcnt(0);
    }
    asm volatile("" ::: "memory");
    const float* buf = (const float*)&lds[(size_t)(s & 1) * STAGE];

    for (int e = lane; e < sCount; e += 32) {
      float x = buf[e];
      if (x > LOGIT_THRESH) {                        // sigmoid(x) > 0.05
        int idx = chunkBase + sBase + e;             // flat index in [C*N)
        int c = idx / N;
        int n = idx - c * N;
        float clsp = sigmoidf_fast(x);
        float cnp  = sigmoidf_fast(cnP[(size_t)b * N + n]);
        float score = clsp * cnp;                    // in (0,1): bits monotone
        unsigned sb = __float_as_uint(score);
        unsigned bin = (sb >> 19) & (BINS - 1);      // exp + 4 mantissa bits
        if (MODE == 0) {
          atomicAdd(&lds[STAGE * 2 + bin], 1u);
        } else if (bin >= cutBin) {
          unsigned pos = atomicAdd(&cnt[b * 3 + level], 1u);
          if (pos < CAP) {
            unsigned* rec = cand + (((size_t)(b * 3 + level)) * CAP + pos) * 2;
            rec[0] = sb;
            rec[1] = ((unsigned)n << 7) | (unsigned)c;  // n<=10000 (14b), c<80 (7b)
          }
        }
      }
    }
  }

  if (MODE == 0) {
    asm volatile("s_wait_dscnt 0" ::: "memory");
    unsigned* gh = hist + (size_t)(b * 3 + level) * BINS;
    for (int i = lane; i < BINS; i += 32) {
      unsigned v = lds[STAGE * 2 + i];
      if (v) atomicAdd(&gh[i], v);
    }
  }
}

// ------------------- kernel 2: histogram suffix-sum cutoff -------------------
__global__ void cutoff_kernel(const unsigned* __restrict__ hist,
                              unsigned* __restrict__ cut) {
  int g = blockIdx.x;  // b*3 + level, 48 groups
  if (threadIdx.x == 0) {
    const unsigned* h = hist + (size_t)g * BINS;
    unsigned acc = 0, c = 0;
    for (int bin = BINS - 1; bin >= 0; --bin) {
      acc += h[bin];
      if (acc >= (unsigned)KTOP) { c = (unsigned)bin; break; }
    }
    cut[g] = c;   // keep bins >= c  ->  >= KTOP candidates (or all if fewer)
  }
}

// deterministic descending order: score desc, key asc on ties
__device__ __forceinline__ bool precede(float sa, unsigned ka, float sb, unsigned kb) {
  return (sa > sb) || (sa == sb && ka < kb);
}

// ------------- kernel 4: per-(image,level) top-K sort + box decode -----------
__global__ __launch_bounds__(512) void topk_decode_kernel(
    const unsigned* __restrict__ cnt, const unsigned* __restrict__ cand,
    const float* __restrict__ loc0, const float* __restrict__ loc1,
    const float* __restrict__ loc2, const float* __restrict__ reg0,
    const float* __restrict__ reg1, const float* __restrict__ reg2,
    float* __restrict__ slab) {
  extern __shared__ unsigned char smem[];
  float*    sc = (float*)smem;                    // CAP
  unsigned* ky = (unsigned*)(smem + (size_t)CAP * 4);
  int g = blockIdx.x;
  int b = g / 3, level = g % 3;
  int count = (int)min(cnt[g], (unsigned)CAP);
  const unsigned* cd = cand + (size_t)g * CAP * 2;

  for (int i = threadIdx.x; i < CAP; i += blockDim.x) {
    if (i < count) { sc[i] = __uint_as_float(cd[i * 2]); ky[i] = cd[i * 2 + 1]; }
    else           { sc[i] = -INFINITY; ky[i] = 0xFFFFFFFFu; }
  }
  __syncthreads();

  for (int k = 2; k <= CAP; k <<= 1) {
    for (int j = k >> 1; j > 0; j >>= 1) {
      for (int t = threadIdx.x; t < CAP; t += blockDim.x) {
        int ixj = t ^ j;
        if (ixj > t) {
          bool desc = ((t & k) == 0);
          float sa = sc[t], sb2 = sc[ixj];
          unsigned ka = ky[t], kb = ky[ixj];
          bool doSwap = desc ? precede(sb2, kb, sa, ka) : precede(sa, ka, sb2, kb);
          if (doSwap) { sc[t] = sb2; sc[ixj] = sa; ky[t] = kb; ky[ixj] = ka; }
        }
      }
      __syncthreads();
    }
  }

  const float* locP = (level == 0) ? loc0 : (level == 1) ? loc1 : loc2;
  const float* regP = (level == 0) ? reg0 : (level == 1) ? reg1 : reg2;
  const int N       = (level == 0) ? N_L0 : (level == 1) ? N_L1 : N_L2;
  float* out = slab + ((size_t)b * TOTC + (size_t)level * KTOP) * 6;

  for (int kk = threadIdx.x; kk < KTOP; kk += blockDim.x) {
    float o0 = 0, o1 = 0, o2 = 0, o3 = 0, oc = 0, os = -INFINITY;
    if (kk < count) {
      float s = sc[kk];
      unsigned key = ky[kk];
      int n = (int)(key >> 7), c = (int)(key & 127u);
      float lx = locP[n * 2], ly = locP[n * 2 + 1];
      size_t rb = (size_t)b * 4 * N;
      float r0 = regP[rb + 0 * (size_t)N + n];
      float r1 = regP[rb + 1 * (size_t)N + n];
      float r2 = regP[rb + 2 * (size_t)N + n];
      float r3 = regP[rb + 3 * (size_t)N + n];
      o0 = fminf(fmaxf(lx - r0, 0.f), IMGMAX);
      o1 = fminf(fmaxf(ly - r1, 0.f), IMGMAX);
      o2 = fminf(fmaxf(lx + r2, 0.f), IMGMAX);
      o3 = fminf(fmaxf(ly + r3, 0.f), IMGMAX);
      oc = (float)(c + 1);
      os = sqrtf(s);               // MIN_SIZE==0 -> always "valid" here
    }
    float* rec = out + (size_t)kk * 6;
    rec[0] = o0; rec[1] = o1; rec[2] = o2; rec[3] = o3; rec[4] = os; rec[5] = oc;
  }
}

// -------------------- kernel 5: per-image NMS + final top-100 ----------------
static constexpr int NMS_OFF_SC   = 0;
static constexpr int NMS_OFF_PID  = NPAD2 * 4;
static constexpr int NMS_OFF_KEEP = NPAD2 * 8;
static constexpr int NMS_OFF_BOX  = NPAD2 * 12;
static constexpr int NMS_BSTRIDE  = 3008 * 4;           // per box-field array
static constexpr int NMS_OFF_OUT  = NMS_OFF_BOX + 5 * NMS_BSTRIDE;
static constexpr int NMS_SHBYTES  = NMS_OFF_OUT + (POSTN + 2) * 4;

__global__ __launch_bounds__(1024) void nms_kernel(
    const float* __restrict__ slab, float* __restrict__ outBoxes,
    float* __restrict__ outScores, int* __restrict__ outCls,
    int* __restrict__ outFinal) {
  extern __shared__ unsigned char smem[];
  float*    sc2  = (float*)(smem + NMS_OFF_SC);
  unsigned* pid  = (unsigned*)(smem + NMS_OFF_PID);
  unsigned* keep = (unsigned*)(smem + NMS_OFF_KEEP);
  float*    bx1  = (float*)(smem + NMS_OFF_BOX);
  float*    by1  = (float*)((unsigned char*)bx1 + NMS_BSTRIDE);
  float*    bx2  = (float*)((unsigned char*)by1 + NMS_BSTRIDE);
  float*    by2  = (float*)((unsigned char*)bx2 + NMS_BSTRIDE);
  float*    bcl  = (float*)((unsigned char*)by2 + NMS_BSTRIDE);
  unsigned* outPos = (unsigned*)(smem + NMS_OFF_OUT);
  unsigned* nKeptP = outPos + POSTN;

  int b = blockIdx.x;
  const float* s = slab + (size_t)b * TOTC * 6;
  for (int i = threadIdx.x; i < TOTC; i += blockDim.x) {
    bx1[i] = s[i * 6 + 0]; by1[i] = s[i * 6 + 1];
    bx2[i] = s[i * 6 + 2]; by2[i] = s[i * 6 + 3];
    bcl[i] = s[i * 6 + 5];
  }
  for (int i = threadIdx.x; i < NPAD2; i += blockDim.x) {
    float v = (i < TOTC) ? s[i * 6 + 4] : -INFINITY;
    sc2[i] = v; pid[i] = (unsigned)i;
  }
  __syncthreads();

  for (int k = 2; k <= NPAD2; k <<= 1) {
    for (int j = k >> 1; j > 0; j >>= 1) {
      for (int t = threadIdx.x; t < NPAD2; t += blockDim.x) {
        int ixj = t ^ j;
        if (ixj > t) {
          bool desc = ((t & k) == 0);
          float sa = sc2[t], sb2 = sc2[ixj];
          unsigned ka = pid[t], kb = pid[ixj];
          bool doSwap = desc ? precede(sb2, kb, sa, ka) : precede(sa, ka, sb2, kb);
          if (doSwap) { sc2[t] = sb2; sc2[ixj] = sa; pid[t] = kb; pid[ixj] = ka; }
        }
      }
      __syncthreads();
    }
  }
  for (int i = threadIdx.x; i < NPAD2; i += blockDim.x)
    keep[i] = (sc2[i] != -INFINITY) ? 1u : 0u;
  __syncthreads();

  // greedy class-agnostic NMS over sorted order (reference +1 IoU convention)
  for (int i = 0; i < TOTC; ++i) {
    float si = sc2[i];
    if (si == -INFINITY) break;          // uniform: remainder are invalid
    if (keep[i]) {
      unsigned pi = pid[i];
      float ax1 = bx1[pi], ay1 = by1[pi], ax2 = bx2[pi], ay2 = by2[pi];
      float aArea = (ax2 - ax1 + 1.f) * (ay2 - ay1 + 1.f);
      for (int j = i + 1 + (int)threadIdx.x; j < TOTC; j += blockDim.x) {
        if (keep[j]) {
          unsigned pj = pid[j];
          float ix1 = fmaxf(ax1, bx1[pj]);
          float iy1 = fmaxf(ay1, by1[pj]);
          float ix2 = fminf(ax2, bx2[pj]);
          float iy2 = fminf(ay2, by2[pj]);
          float inter = fmaxf(ix2 - ix1 + 1.f, 0.f) * fmaxf(iy2 - iy1 + 1.f, 0.f);
          float bArea = (bx2[pj] - bx1[pj] + 1.f) * (by2[pj] - by1[pj] + 1.f);
          if (inter / (aArea + bArea - inter) > NMS_IOU) keep[j] = 0u;
        }
      }
    }
    __syncthreads();
  }

  if (threadIdx.x == 0) {
    unsigned nk = 0;
    for (int i = 0; i < TOTC && nk < POSTN; ++i) {
      if (sc2[i] == -INFINITY) break;
      if (keep[i]) outPos[nk++] = (unsigned)i;
    }
    *nKeptP = nk;
  }
  __syncthreads();
  unsigned nk = *nKeptP;

  for (int r = threadIdx.x; r < POSTN; r += blockDim.x) {
    float o0 = 0, o1 = 0, o2 = 0, o3 = 0, s4 = 0;
    int oc = 0, fin = 0;
    if (r < nk) {
      int p = (int)outPos[r];
      float sv = sc2[p];
      if (sv >= PRE_THRESH) {            // final score filter
        unsigned pi2 = pid[p];
        o0 = bx1[pi2]; o1 = by1[pi2]; o2 = bx2[pi2]; o3 = by2[pi2];
        s4 = sv; oc = (int)bcl[pi2]; fin = 1;
      }
    }
    float* ob = outBoxes + ((size_t)b * POSTN + r) * 4;
    ob[0] = o0; ob[1] = o1; ob[2] = o2; ob[3] = o3;
    outScores[b * POSTN + r] = s4;
    outCls[b * POSTN + r]    = oc;
    outFinal[b * POSTN + r]  = fin;
  }
}

// ----------------------------------------------------------------------------
extern "C" void kernel_launch(void* const* d_in, const int* in_sizes, int n_in,
                              void* d_out, int out_size, void* d_ws,
                              size_t ws_size, hipStream_t stream) {
  (void)in_sizes; (void)n_in; (void)out_size; (void)ws_size;
  const float* loc0 = (const float*)d_in[0];
  const float* cls0 = (const float*)d_in[1];
  const float* reg0 = (const float*)d_in[2];
  const float* cn0  = (const float*)d_in[3];
  const float* loc1 = (const float*)d_in[4];
  const float* cls1 = (const float*)d_in[5];
  const float* reg1 = (const float*)d_in[6];
  const float* cn1  = (const float*)d_in[7];
  const float* loc2 = (const float*)d_in[8];
  const float* cls2 = (const float*)d_in[9];
  const float* reg2 = (const float*)d_in[10];
  const float* cn2  = (const float*)d_in[11];

  unsigned* ws   = (unsigned*)d_ws;
  unsigned* hist = ws + WS_HIST;
  unsigned* cut  = ws + WS_CUT;
  unsigned* cnt  = ws + WS_CNT;
  unsigned* cand = ws + WS_CAND;
  float*    slab = (float*)(ws + WS_SLAB);

  float* outBoxes  = (float*)d_out;
  float* outScores = outBoxes + (size_t)BATCH * POSTN * 4;
  int*   outCls    = (int*)(outScores + (size_t)BATCH * POSTN);
  int*   outFinal  = outCls + (size_t)BATCH * POSTN;

  (void)hipFuncSetAttribute((const void*)topk_decode_kernel,
                            hipFuncAttributeMaxDynamicSharedMemorySize, CAP * 8);
  (void)hipFuncSetAttribute((const void*)nms_kernel,
                            hipFuncAttributeMaxDynamicSharedMemorySize, NMS_SHBYTES);

  zero_ws_kernel<<<(WS_ZERO_WORDS + 255) / 256, 256, 0, stream>>>(ws, WS_ZERO_WORDS);
  scan_kernel<0><<<BATCH * CHTOT, 32, 0, stream>>>(cls0, cls1, cls2, cn0, cn1,
                                                   cn2, hist, cut, cnt, cand);
  cutoff_kernel<<<BATCH * 3, 32, 0, stream>>>(hist, cut);
  scan_kernel<1><<<BATCH * CHTOT, 32, 0, stream>>>(cls0, cls1, cls2, cn0, cn1,
                                                   cn2, hist, cut, cnt, cand);
  topk_decode_kernel<<<BATCH * 3, 512, CAP * 8, stream>>>(
      cnt, cand, loc0, loc1, loc2, reg0, reg1, reg2, slab);
  nms_kernel<<<BATCH, 1024, NMS_SHBYTES, stream>>>(slab, outBoxes, outScores,
                                                   outCls, outFinal);
}